// baseIGNN_30064771072230
// MI455X (gfx1250) — compile-verified
//
#include <hip/hip_runtime.h>
#include <hip/hip_bf16.h>

typedef float v2f __attribute__((ext_vector_type(2)));
typedef float v8f __attribute__((ext_vector_type(8)));

#define DINC 128   // in_channels == out_channels == 128

// ---------------- degree / norm kernels ----------------

__global__ void k_init_deg(float* __restrict__ deg, int n) {
    int i = blockIdx.x * blockDim.x + threadIdx.x;
    if (i < n) deg[i] = 1.0f;              // self-loop contributes 1
}

__global__ void k_count_deg(const int* __restrict__ col, float* __restrict__ deg, int e) {
    int i = blockIdx.x * blockDim.x + threadIdx.x;
    if (i < e) atomicAdd(&deg[col[i]], 1.0f);
}

__global__ void k_rsqrt(const float* __restrict__ deg, float* __restrict__ dis, int n) {
    int i = blockIdx.x * blockDim.x + threadIdx.x;
    if (i < n) dis[i] = rsqrtf(deg[i]);    // deg >= 1 always
}

// ---------------- fp32 WMMA GEMM: out[M][128] = A[M][128] @ W[128][128]^T ----
//
// One wave32 computes a 16x128 output strip (8 tiles of 16x16) so the A
// fragment is reused 8x from registers and the 8 accumulator chains are
// independent (WMMA pipe stays busy; dependent WMMAs are 8 issues apart).
//
// W is staged once per block into 64KB of LDS, transposed into float2 pairs:
//   lw2[kp*128 + n] = { W[n][2kp], W[n][2kp+1] },  kp in [0,64)
// so the B fragment for V_WMMA_F32_16X16X4_F32 is one ds_load_b64 with lanes
// reading consecutive float2s (conflict-light).
//
// Hardware fragment layouts (ISA 7.12.2, fp32 16x16x4):
//   A: lanes 0-15 rows m0..m0+15 with K=k0,k0+1; lanes 16-31 K=k0+2,k0+3
//   B: mirrors with N across lanes
//   C/D: VGPR r -> row m0+r (lanes<16) / m0+r+8 (lanes>=16), col n0+(lane&15)

__global__ void k_gemm_wmma_f32(const float* __restrict__ A,
                                const float* __restrict__ W,   // [128][128] row-major
                                float* __restrict__ out,
                                int M) {
    constexpr int K = DINC, NC = DINC;
    __shared__ float2 lw2[64 * 128];                   // exactly 64 KB

    const int lane = threadIdx.x & 31;
    const int wave = threadIdx.x >> 5;                 // 8 waves / block
    const int m0   = (blockIdx.x * 8 + wave) * 16;     // 128 rows per block
    const int half = lane >> 4;                        // K sub-pair select
    const int l15  = lane & 15;

    int rowA = m0 + l15;
    if (rowA >= M) rowA = M - 1;                       // clamp (M % 16 == 0 anyway)
    const float2* __restrict__ Av = (const float2*)(A + (size_t)rowA * K);

    // pull this wave's A rows toward the WGP while we stage W into LDS
    __builtin_prefetch(Av, 0, 0);
    __builtin_prefetch(((const char*)Av) + 256, 0, 0);

    // stage W -> LDS transposed (writes are LDS-conflict-free; W is 64KB and
    // L2-hot, so the strided global pattern is cheap)
    const float2* __restrict__ Wv = (const float2*)W;  // global idx n*64 + kp
    for (int idx = threadIdx.x; idx < 64 * 128; idx += 256) {
        const int kp = idx >> 7;
        const int n  = idx & 127;
        lw2[idx] = Wv[n * 64 + kp];                    // lw2[kp*128 + n]
    }
    __syncthreads();

    if (m0 >= M) return;                               // wave-uniform; EXEC all-ones below

    v8f acc[8];
    #pragma unroll
    for (int t = 0; t < 8; ++t) acc[t] = (v8f){0.f,0.f,0.f,0.f,0.f,0.f,0.f,0.f};

    for (int k0 = 0; k0 < K; k0 += 4) {
        const int kp = (k0 >> 1) + half;               // ka = 2*kp
        const float2 av = Av[kp];
        v2f a; a.x = av.x; a.y = av.y;
        const float2* __restrict__ brow = &lw2[kp * 128 + l15];
        #pragma unroll
        for (int t = 0; t < 8; ++t) {
            const float2 bv = brow[t * 16];
            v2f b; b.x = bv.x; b.y = bv.y;
            // (neg_a, A, neg_b, B, c_mod, C, reuse_a, reuse_b)
            acc[t] = __builtin_amdgcn_wmma_f32_16x16x4_f32(false, a, false, b,
                                                           (short)0, acc[t],
                                                           false, false);
        }
    }

    #pragma unroll
    for (int t = 0; t < 8; ++t) {
        float* o = out + (size_t)(m0 + half * 8) * NC + t * 16 + l15;
        #pragma unroll
        for (int r = 0; r < 8; ++r) {
            if (m0 + half * 8 + r < M) o[(size_t)r * NC] = acc[t][r];
        }
    }
}

// ---------------- edge scatter: out[col] += x[row] * dis[row]*dis[col] -------
// One wave32 per edge; 32 lanes x float4 = 128 floats per row. x and out are
// L2-resident (51MB each vs 192MB L2) so this is L2-atomic-throughput bound.

__global__ void k_scatter(const int* __restrict__ rows, const int* __restrict__ cols,
                          const float* __restrict__ x, const float* __restrict__ dis,
                          float* __restrict__ out, int e) {
    const int wid  = (blockIdx.x * blockDim.x + threadIdx.x) >> 5;
    const int lane = threadIdx.x & 31;
    if (wid >= e) return;
    const int r = rows[wid];
    const int c = cols[wid];
    const float norm = dis[r] * dis[c];
    const float4 v = ((const float4*)(x + (size_t)r * DINC))[lane];
    float* o = out + (size_t)c * DINC + lane * 4;
    atomicAdd(&o[0], v.x * norm);
    atomicAdd(&o[1], v.y * norm);
    atomicAdd(&o[2], v.z * norm);
    atomicAdd(&o[3], v.w * norm);
}

// ---------------- self-loop + ReLU: out = relu(out + x * dis^2) --------------

__global__ void k_selfloop_relu(const float* __restrict__ x, const float* __restrict__ dis,
                                float* __restrict__ out, long total) {
    long i = (long)blockIdx.x * blockDim.x + threadIdx.x;
    if (i < total) {
        const int node = (int)(i >> 7);                // /128
        const float d = dis[node];
        float v = out[i] + x[i] * d * d;
        out[i] = v > 0.f ? v : 0.f;
    }
}

// ---------------- launch ----------------

extern "C" void kernel_launch(void* const* d_in, const int* in_sizes, int n_in,
                              void* d_out, int out_size, void* d_ws, size_t ws_size,
                              hipStream_t stream) {
    const float* feature   = (const float*)d_in[0];   // [N,128]
    const int*   edgeIndex = (const int*)  d_in[1];   // [2,E]
    const float* embedding = (const float*)d_in[2];   // [N,128]
    const float* W_conv    = (const float*)d_in[3];   // [128,128]
    const float* W_mlp     = (const float*)d_in[4];   // [128,128]
    float*       out       = (float*)d_out;           // [N,128]

    const int N = in_sizes[0] / DINC;
    const int E = in_sizes[1] / 2;
    const int* rows = edgeIndex;
    const int* cols = edgeIndex + E;

    // workspace layout: x [N*128] | deg [N] | dis [N]
    float* x   = (float*)d_ws;
    float* deg = x + (size_t)N * DINC;
    float* dis = deg + N;

    // 1) deg = 1 (self-loops)
    k_init_deg<<<(N + 255) / 256, 256, 0, stream>>>(deg, N);
    // 2) deg[col] += 1 per edge
    k_count_deg<<<(E + 255) / 256, 256, 0, stream>>>(cols, deg, E);
    // 3) dis = rsqrt(deg)
    k_rsqrt<<<(N + 255) / 256, 256, 0, stream>>>(deg, dis, N);

    // 4) x = embedding @ W_conv^T   (WMMA fp32, 128 rows per block)
    const int gblk = (N + 127) / 128;
    k_gemm_wmma_f32<<<gblk, 256, 0, stream>>>(embedding, W_conv, x, N);

    // 5) out = feature @ W_mlp^T    (WMMA fp32; full overwrite of d_out)
    k_gemm_wmma_f32<<<gblk, 256, 0, stream>>>(feature, W_mlp, out, N);

    // 6) out[col] += x[row] * dis[row]*dis[col]  (one wave per edge)
    const int sblk = (E + 7) / 8;                      // 8 waves per block
    k_scatter<<<sblk, 256, 0, stream>>>(rows, cols, x, dis, out, E);

    // 7) out = relu(out + x * dis^2)
    const long total = (long)N * DINC;
    k_selfloop_relu<<<(int)((total + 255) / 256), 256, 0, stream>>>(x, dis, out, total);
}